// SpectralConv3d_1340029796994
// MI455X (gfx1250) — compile-verified
//
#include <hip/hip_runtime.h>

typedef __attribute__((ext_vector_type(2))) float v2f;
typedef __attribute__((ext_vector_type(8))) float v8f;

#if defined(__HIP_DEVICE_COMPILE__)
#if !__has_builtin(__builtin_amdgcn_wmma_f32_16x16x4_f32)
#error "missing __builtin_amdgcn_wmma_f32_16x16x4_f32 on this toolchain"
#endif
#endif

// CDNA5 async global->LDS staging (ASYNCcnt-tracked) when available.
#if defined(__HIP_DEVICE_COMPILE__) && \
    __has_builtin(__builtin_amdgcn_global_load_async_to_lds_b32) && \
    __has_builtin(__builtin_amdgcn_s_wait_asynccnt)
#define ASYNC_LDS 1
#else
#define ASYNC_LDS 0
#endif

#define LDSC  72   // padded LDS row pitch (floats)
#define KMAXR 68   // max padded K (z-axis rfft: K=65 -> 68)

typedef __attribute__((address_space(1))) int as1_int;
typedef __attribute__((address_space(3))) int as3_int;

__device__ __forceinline__ void stage_word(const float* __restrict__ g,
                                           float* l, bool valid) {
  if (valid) {
#if ASYNC_LDS
    __builtin_amdgcn_global_load_async_to_lds_b32(
        (as1_int*)(void*)const_cast<float*>(g),
        (as3_int*)(void*)l, 0, 0);
#else
    *l = *g;
#endif
  } else {
    *l = 0.f;
  }
}

__device__ __forceinline__ long line_off(int l, int d0, int d1,
                                         long s0, long s1, long s2) {
  int h0 = l % d0;
  int r  = l / d0;
  int h1 = r % d1;
  int h2 = r / d1;
  return (long)h0 * s0 + (long)h1 * s1 + (long)h2 * s2;
}

// ---------------------------------------------------------------------------
// Fused spectral "lines GEMM" on V_WMMA_F32_16X16X4_F32.
//   MODE 0 (r2c): OutRe = A*Mc,           OutIm = A*Ms
//   MODE 1 (c2c): OutRe = Ar*Mc - Ai*Ms,  OutIm = Ar*Ms + Ai*Mc
//   MODE 2 (c2r): OutRe = Ar*Mc + Ai*Ms
// Each block: 128 threads = 4 waves, 4 wave-tiles of 16 lines (64 lines).
// Line l = (h2*d1 + h1)*d0 + h0 with per-component in/out strides.
// ---------------------------------------------------------------------------
template <int MODE>
__global__ void gemm_spec(const float* __restrict__ InRe,
                          const float* __restrict__ InIm,
                          const float* __restrict__ Mc,
                          const float* __restrict__ Ms,
                          float* __restrict__ OutRe,
                          float* __restrict__ OutIm,
                          int L, int K, int N, int d0, int d1,
                          long si0, long si1, long si2, long ska,
                          long so0, long so1, long so2, long sn) {
  __shared__ float sA[(MODE == 0) ? 1 : 2][64][LDSC];
  __shared__ float sM[2][KMAXR][LDSC];

  const int tid   = threadIdx.x;
  const int K4    = (K + 3) & ~3;
  const int lbase = blockIdx.x * 64;

  // Stage the two DFT matrices (cos / sin planes) into LDS, zero-padded.
  for (int idx = tid; idx < K4 * LDSC; idx += 128) {
    int k = idx / LDSC, j = idx % LDSC;
    bool v = (k < K && j < N);
    long o = v ? (long)k * N + j : 0;
    stage_word(Mc + o, &sM[0][k][j], v);
    stage_word(Ms + o, &sM[1][k][j], v);
  }
  // Stage 64 input lines (re, and im for complex modes).
  for (int idx = tid; idx < 64 * LDSC; idx += 128) {
    int r = idx / LDSC, k = idx % LDSC;
    int l = lbase + r;
    bool v = (l < L && k < K);
    long off = v ? line_off(l, d0, d1, si0, si1, si2) + (long)k * ska : 0;
    stage_word(InRe + off, &sA[0][r][k], v);
    if (MODE != 0) stage_word(InIm + off, &sA[(MODE == 0) ? 0 : 1][r][k], v);
  }
#if ASYNC_LDS
  __builtin_amdgcn_s_wait_asynccnt(0);
#endif
  __syncthreads();

  const int wv = tid >> 5;
  const int ln = tid & 31;
  const int m  = ln & 15;          // A row within tile / output column
  const int hi = ln >> 4;          // half-wave: K pair 0/1 vs 2/3
  const int arow = wv * 16 + m;    // this wave's A row in LDS
  const int nt = (N + 15) >> 4;

  for (int jt = 0; jt < nt; ++jt) {
    const int jb = jt << 4;
    v8f t1 = {0.f, 0.f, 0.f, 0.f, 0.f, 0.f, 0.f, 0.f};
    v8f t2 = t1, t3 = t1, t4 = t1;
    for (int k = 0; k < K4; k += 4) {
      v2f ar, ai, bc, bs;
      // A 16x4 f32 frag: lanes 0-15 K=k..k+1, lanes 16-31 K=k+2..k+3
      ar.x = sA[0][arow][k + 2 * hi];
      ar.y = sA[0][arow][k + 2 * hi + 1];
      // B 4x16 f32 frags from cos/sin matrices
      bc.x = sM[0][k + 2 * hi][jb + m];
      bc.y = sM[0][k + 2 * hi + 1][jb + m];
      bs.x = sM[1][k + 2 * hi][jb + m];
      bs.y = sM[1][k + 2 * hi + 1][jb + m];
      if (MODE != 0) {
        ai.x = sA[(MODE == 0) ? 0 : 1][arow][k + 2 * hi];
        ai.y = sA[(MODE == 0) ? 0 : 1][arow][k + 2 * hi + 1];
      }
      if (MODE == 0) {
        t1 = __builtin_amdgcn_wmma_f32_16x16x4_f32(false, ar, false, bc, (short)0, t1, false, false);
        t2 = __builtin_amdgcn_wmma_f32_16x16x4_f32(false, ar, false, bs, (short)0, t2, false, false);
      } else if (MODE == 1) {
        t1 = __builtin_amdgcn_wmma_f32_16x16x4_f32(false, ar, false, bc, (short)0, t1, false, false);
        t2 = __builtin_amdgcn_wmma_f32_16x16x4_f32(false, ai, false, bs, (short)0, t2, false, false);
        t3 = __builtin_amdgcn_wmma_f32_16x16x4_f32(false, ar, false, bs, (short)0, t3, false, false);
        t4 = __builtin_amdgcn_wmma_f32_16x16x4_f32(false, ai, false, bc, (short)0, t4, false, false);
      } else {
        t1 = __builtin_amdgcn_wmma_f32_16x16x4_f32(false, ar, false, bc, (short)0, t1, false, false);
        t1 = __builtin_amdgcn_wmma_f32_16x16x4_f32(false, ai, false, bs, (short)0, t1, false, false);
      }
    }
    const int n = jb + m;
    if (n < N) {
#pragma unroll
      for (int v = 0; v < 8; ++v) {
        int row = v + 8 * hi;                  // 16x16 f32 C/D layout
        int l   = lbase + wv * 16 + row;
        if (l < L) {
          long o = line_off(l, d0, d1, so0, so1, so2) + (long)n * sn;
          if (MODE == 0) {
            OutRe[o] = t1[v];
            OutIm[o] = t2[v];
          } else if (MODE == 1) {
            OutRe[o] = t1[v] - t2[v];
            OutIm[o] = t3[v] + t4[v];
          } else {
            OutRe[o] = t1[v];
          }
        }
      }
    }
  }
}

// ---------------------------------------------------------------------------
// M[k][n] = scale * w_k * trig(sgn * 2pi * k * n / P);  dcw: w_0=1, w_k=2.
// ---------------------------------------------------------------------------
__global__ void fill_trig(float* __restrict__ M, int K, int N, int P,
                          float sgn, float scale, int mode, int dcw) {
  int t = blockIdx.x * blockDim.x + threadIdx.x;
  if (t >= K * N) return;
  int k = t / N, n = t % N;
  float w   = (dcw && k > 0) ? 2.f : 1.f;
  int   kn  = (k * n) % P;
  float ang = sgn * 6.28318530717958647692f * (float)kn / (float)P;
  float v   = mode ? __sinf(ang) : __cosf(ang);
  M[t] = scale * w * v;
}

__global__ void k_zero(float* __restrict__ p, long n) {
  long t = (long)blockIdx.x * blockDim.x + threadIdx.x;
  if (t < n) p[t] = 0.f;
}

__global__ void k_hmult(const float* __restrict__ gr, const float* __restrict__ gi,
                        float* __restrict__ fr, float* __restrict__ fi, long n) {
  long t = (long)blockIdx.x * blockDim.x + threadIdx.x;
  if (t >= n) return;
  int kz = (int)(t % 33);
  float h = (kz >= 1 && kz <= 16) ? 2.f : 0.f;
  fr[t] = gr[t] * h;
  fi[t] = gi[t] * h;
}

__global__ void k_gather_hart(const float* __restrict__ sre,
                              const float* __restrict__ sim,
                              float* __restrict__ xr, long n) {
  long t = (long)blockIdx.x * blockDim.x + threadIdx.x;
  if (t >= n) return;
  int p = (int)(t & 4095);
  int q = (int)(t >> 12);
  int blk = q % 3; q /= 3;
  int ch  = q & 31;
  int b   = q >> 5;
  int lx = p >> 8, ly = (p >> 4) & 15, lz = p & 15;
  int kx = (blk == 1) ? 48 + lx : lx;
  int ky = (blk == 2) ? 48 + ly : ly;
  long idx = (long)b * 4325376 + (long)ch * 135168 + (long)kx * 2112 +
             (long)ky * 33 + lz;
  xr[t] = sre[idx] - sim[idx];
}

__global__ void k_gather_hilb(const float* __restrict__ hil,
                              float* __restrict__ xi, long n) {
  long t = (long)blockIdx.x * blockDim.x + threadIdx.x;
  if (t >= n) return;
  int p = (int)(t & 4095);
  int q = (int)(t >> 12);
  int blk = q % 3; q /= 3;
  int ch  = q & 31;
  int b   = q >> 5;
  int lx = p >> 8, ly = (p >> 4) & 15, lz = p & 15;
  int sx = (blk == 1) ? 48 + lx : lx;
  int sy = (blk == 2) ? 48 + ly : ly;
  long idx = (long)b * 2097152 + (long)ch * 65536 + (long)sx * 1024 +
             (long)sy * 16 + lz;
  xi[t] = hil[idx];
}

// compl_mul3d real part:  0.5*Sum_i [ x1*(w + wn) + x1n*(w - wn) ]
__global__ void k_modemul(const float* __restrict__ xre,
                          const float* __restrict__ xim,
                          const float* __restrict__ w1r, const float* __restrict__ w1i,
                          const float* __restrict__ w2r, const float* __restrict__ w2i,
                          const float* __restrict__ w3r, const float* __restrict__ w3i,
                          float* __restrict__ outc, long n) {
  long t = (long)blockIdx.x * blockDim.x + threadIdx.x;
  if (t >= n) return;
  int p = (int)(t & 4095);
  int q = (int)(t >> 12);
  int blk = q % 3; q /= 3;
  int o   = q & 31;
  int b   = q >> 5;
  const float* wr = (blk == 0) ? w1r : (blk == 1) ? w2r : w3r;
  const float* wi = (blk == 0) ? w1i : (blk == 1) ? w2i : w3i;
  int lx = p >> 8, ly = (p >> 4) & 15, lz = p & 15;
  int np = (((16 - lx) & 15) << 8) | (((16 - ly) & 15) << 4) | ((16 - lz) & 15);
  float acc = 0.f;
  for (int i = 0; i < 32; ++i) {
    long xb = ((long)((b * 32 + i) * 3 + blk)) << 12;
    long wb = ((long)(i * 32 + o)) << 12;
    float ar  = xre[xb + p],  ai  = xim[xb + p];
    float anr = xre[xb + np], ani = xim[xb + np];
    float wrp = wr[wb + p],  wrn = wr[wb + np];
    float wip = wi[wb + p],  win = wi[wb + np];
    acc += ar * (wrp + wrn) - ai * (wip + win)
         + anr * (wrp - wrn) - ani * (wip - win);
  }
  outc[t] = 0.5f * acc;
}

__global__ void k_scatter(const float* __restrict__ outc,
                          float* __restrict__ R, long n) {
  long t = (long)blockIdx.x * blockDim.x + threadIdx.x;
  if (t >= n) return;
  int p = (int)(t & 4095);
  int q = (int)(t >> 12);
  int blk = q % 3; q /= 3;
  int o   = q & 31;
  int b   = q >> 5;
  int lx = p >> 8, ly = (p >> 4) & 15, lz = p & 15;
  int gx = (blk == 1) ? 48 + lx : lx;
  int gy = (blk == 2) ? 48 + ly : ly;
  R[(long)b * 4325376 + (long)o * 135168 + (long)gx * 2112 + (long)gy * 33 + lz] =
      outc[t];
}

__global__ void k_combine(const float* __restrict__ fre,
                          const float* __restrict__ fim,
                          float* __restrict__ out, long n) {
  long t = (long)blockIdx.x * blockDim.x + threadIdx.x;
  if (t >= n) return;
  int kz = (int)(t % 33);
  long q = t / 33;
  int y = (int)(q % 64); q /= 64;
  int x = (int)(q % 64); q /= 64;
  int o = (int)(q % 32);
  int b = (int)(q / 32);
  const float inv = 1.f / 17301504.f;   // 4*32*64*64*33
  float v;
  if (kz <= 16) {
    long idx = (long)b * 2228224 + (long)o * 69632 + (long)x * 1088 +
               (long)y * 17 + kz;
    v = fre[idx] - fim[idx];
  } else {
    long idx = (long)b * 2228224 + (long)(31 - o) * 69632 +
               (long)(63 - x) * 1088 + (long)(63 - y) * 17 + (32 - kz);
    v = fre[idx] + fim[idx];
  }
  out[t] = v * inv;
}

// ---------------------------------------------------------------------------
extern "C" void kernel_launch(void* const* d_in, const int* in_sizes, int n_in,
                              void* d_out, int out_size, void* d_ws, size_t ws_size,
                              hipStream_t stream) {
  const float* x   = (const float*)d_in[0];
  const float* w1r = (const float*)d_in[1];
  const float* w1i = (const float*)d_in[2];
  const float* w2r = (const float*)d_in[3];
  const float* w2i = (const float*)d_in[4];
  const float* w3r = (const float*)d_in[5];
  const float* w3i = (const float*)d_in[6];
  float* out = (float*)d_out;
  float* ws  = (float*)d_ws;

  const long SZ1 = 17301504L;   // B*C*64*64*33
  const long SZc = 1572864L;    // B*C*3*4096

  float* MAT = ws;              // 14 matrix slots * 4096 floats
  float* Fre = ws + 65536;
  float* Fim = Fre + SZ1;
  float* Gre = Fim + SZ1;
  float* Gim = Gre + SZ1;
  float* xcr = Gim + SZ1;
  float* xci = xcr + SZc;
  float* oc  = xci + SZc;
  (void)ws_size; (void)in_sizes; (void)n_in; (void)out_size;

  auto M = [&](int i) { return MAT + (long)i * 4096; };
  auto fill = [&](int slot, int K, int N, int P, float sgn, float sc, int mode, int dcw) {
    int n = K * N;
    fill_trig<<<(n + 255) / 256, 256, 0, stream>>>(M(slot), K, N, P, sgn, sc, mode, dcw);
  };
  fill(0, 65, 33, 65,  1.f, 1.f, 0, 0);          // z fwd cos
  fill(1, 65, 33, 65, -1.f, 1.f, 1, 0);          // z fwd sin
  fill(2, 64, 64, 64,  1.f, 1.f, 0, 0);          // 64 fwd cos
  fill(3, 64, 64, 64, -1.f, 1.f, 1, 0);          // 64 fwd sin
  fill(4, 32, 32, 32,  1.f, 1.f, 0, 0);          // 32 fwd cos
  fill(5, 32, 32, 32, -1.f, 1.f, 1, 0);          // 32 fwd sin
  fill(6, 32, 32, 32,  1.f, 1.f / 32.f, 0, 0);   // 32 inv cos
  fill(7, 32, 32, 32,  1.f, 1.f / 32.f, 1, 0);   // 32 inv sin
  fill(8, 64, 64, 64,  1.f, 1.f / 64.f, 0, 0);   // 64 inv cos
  fill(9, 64, 64, 64,  1.f, 1.f / 64.f, 1, 0);   // 64 inv sin
  fill(10, 33, 16, 65,  1.f, 1.f / 65.f, 0, 1);  // z irfft cos (w_k)
  fill(11, 33, 16, 65, -1.f, 1.f / 65.f, 1, 1);  // z irfft -sin (w_k)
  fill(12, 33, 17, 33,  1.f, 1.f, 0, 0);         // z' fwd cos (len 33)
  fill(13, 33, 17, 33, -1.f, 1.f, 1, 0);         // z' fwd sin

  // ---- forward rfftn over (c,x,y,z) ----
  gemm_spec<0><<<8192, 128, 0, stream>>>(x, x, M(0), M(1), Fre, Fim,
      524288, 65, 33, 64, 64, 65, 4160, 266240, 1, 33, 2112, 135168, 1);
  gemm_spec<1><<<8448, 128, 0, stream>>>(Fre, Fim, M(4), M(5), Gre, Gim,
      540672, 32, 32, 135168, 4, 1, 4325376, 0, 135168, 1, 4325376, 0, 135168);
  gemm_spec<1><<<4224, 128, 0, stream>>>(Gre, Gim, M(2), M(3), Fre, Fim,
      270336, 64, 64, 2112, 128, 1, 135168, 0, 2112, 1, 135168, 0, 2112);
  gemm_spec<1><<<4224, 128, 0, stream>>>(Fre, Fim, M(2), M(3), Gre, Gim,
      270336, 64, 64, 33, 64, 1, 2112, 135168, 33, 1, 2112, 135168, 33);

  // Hartley corners (dht part of xc); spectrum lives in G.
  k_gather_hart<<<6144, 256, 0, stream>>>(Gre, Gim, xcr, SZc);

  // ---- hilbert: h-mask then 4-axis inverse ----
  k_hmult<<<67584, 256, 0, stream>>>(Gre, Gim, Fre, Fim, SZ1);
  gemm_spec<1><<<8448, 128, 0, stream>>>(Fre, Fim, M(6), M(7), Gre, Gim,
      540672, 32, 32, 135168, 4, 1, 4325376, 0, 135168, 1, 4325376, 0, 135168);
  gemm_spec<1><<<4224, 128, 0, stream>>>(Gre, Gim, M(8), M(9), Fre, Fim,
      270336, 64, 64, 2112, 128, 1, 135168, 0, 2112, 1, 135168, 0, 2112);
  gemm_spec<1><<<4224, 128, 0, stream>>>(Fre, Fim, M(8), M(9), Gre, Gim,
      270336, 64, 64, 33, 64, 1, 2112, 135168, 33, 1, 2112, 135168, 33);
  // z inverse (half-spectrum irfft, z=0..15 only): G -> real Hil in Fre
  gemm_spec<2><<<8192, 128, 0, stream>>>(Gre, Gim, M(10), M(11), Fre, Fre,
      524288, 33, 16, 64, 64, 33, 2112, 135168, 1, 16, 1024, 65536, 1);
  k_gather_hilb<<<6144, 256, 0, stream>>>(Fre, xci, SZc);

  // ---- mode mixing (channel contraction, real part only; L2-resident) ----
  k_modemul<<<6144, 256, 0, stream>>>(xcr, xci, w1r, w1i, w2r, w2i, w3r, w3i, oc, SZc);

  // ---- scatter into R = real(out) spectrum, then final 4-axis DHT ----
  k_zero<<<67584, 256, 0, stream>>>(Fre, SZ1);                 // R lives in Fre
  k_scatter<<<6144, 256, 0, stream>>>(oc, Fre, SZc);
  gemm_spec<0><<<8192, 128, 0, stream>>>(Fre, Fre, M(12), M(13), Gre, Gim,
      524288, 33, 17, 64, 64, 33, 2112, 135168, 1, 17, 1088, 69632, 1);
  gemm_spec<1><<<4352, 128, 0, stream>>>(Gre, Gim, M(4), M(5), Fre, Fim,
      278528, 32, 32, 69632, 4, 1, 2228224, 0, 69632, 1, 2228224, 0, 69632);
  gemm_spec<1><<<2176, 128, 0, stream>>>(Fre, Fim, M(2), M(3), Gre, Gim,
      139264, 64, 64, 1088, 128, 1, 69632, 0, 1088, 1, 69632, 0, 1088);
  gemm_spec<1><<<2176, 128, 0, stream>>>(Gre, Gim, M(2), M(3), Fre, Fim,
      139264, 64, 64, 17, 64, 1, 1088, 69632, 17, 1, 1088, 69632, 17);
  k_combine<<<67584, 256, 0, stream>>>(Fre, Fim, out, 17301504L);
}